// udkws_aclap_ma_29910152249995
// MI455X (gfx1250) — compile-verified
//
#include <hip/hip_runtime.h>
#include <math.h>

typedef float v2f __attribute__((ext_vector_type(2)));
typedef float v8f __attribute__((ext_vector_type(8)));
typedef int   v4i __attribute__((vector_size(16)));

typedef __attribute__((address_space(1))) v4i* gv4i_p;   // global int4*
typedef __attribute__((address_space(3))) v4i* lv4i_p;   // LDS int4*

#define LDSS 388          // padded LDS row stride: %64==4 (bank-conflict-free cols), %4==0 (16B rows)
#define DOUT 128
#define NT   4            // M-tiles (of 16 rows) per block
#define THTA 0.8f
#define L2_02 (-2.3219280948873623f)   // log2(0.2)

// ---------------------------------------------------------------------------
// Async global -> LDS staging helpers (CDNA5 GLOBAL_LOAD_ASYNC_TO_LDS_B128)
// ---------------------------------------------------------------------------
__device__ __forceinline__ void async_copy_b128(const float* gsrc, float* ldst) {
#if defined(__has_builtin) && __has_builtin(__builtin_amdgcn_global_load_async_to_lds_b128)
    __builtin_amdgcn_global_load_async_to_lds_b128(
        (gv4i_p)gsrc, (lv4i_p)ldst, 0, 0);
#else
    // fallback: plain 16B copy through VGPRs
    float4 v = *(const float4*)gsrc;
    *(float4*)ldst = v;
#endif
}

__device__ __forceinline__ void wait_async0() {
#if defined(__has_builtin) && __has_builtin(__builtin_amdgcn_s_wait_asynccnt)
    __builtin_amdgcn_s_wait_asynccnt(0);
#else
    asm volatile("s_wait_asynccnt 0x0" ::: "memory");
#endif
}

// ---------------------------------------------------------------------------
// Weight transpose: W[O][I] -> Wt[I][O]
// ---------------------------------------------------------------------------
__global__ void transpose_kernel(const float* __restrict__ W, float* __restrict__ Wt,
                                 int O, int I) {
    int g = blockIdx.x * blockDim.x + threadIdx.x;
    if (g < O * I) {
        int o = g / I, i = g % I;
        Wt[(size_t)i * O + o] = W[g];
    }
}

// ---------------------------------------------------------------------------
// Fused LayerNorm -> GEMM(Kdim -> 128) -> SiLU.
// Each block: 64 rows = NT(4) tiles of 16. 256 threads = 8 waves; wave w owns
// the 16-wide N tile at N=16*w and keeps NT accumulators so each B fragment
// (loaded once per k-step) is reused across all 4 M tiles.
// x tiles staged to LDS via async DMA, then normalized in place.
// ---------------------------------------------------------------------------
__global__ __launch_bounds__(256)
void stage_kernel(const float* __restrict__ x, const float* __restrict__ g,
                  const float* __restrict__ b, const float* __restrict__ Wt,
                  const float* __restrict__ wb, float* __restrict__ y, int Kdim) {
    extern __shared__ float sx[];   // NT * 16 * LDSS floats (dynamic)

    const int tid  = threadIdx.x;
    const int row0 = blockIdx.x * (16 * NT);

    // ---- stage all NT tiles into LDS via async DMA (16B granules) ----
    {
        const int nvec = Kdim >> 2;            // float4s per row
        for (int t = 0; t < NT; t++) {
            for (int e = tid; e < 16 * nvec; e += 256) {
                const int row = e / nvec, c4 = (e % nvec) << 2;
                async_copy_b128(x + (size_t)(row0 + t * 16 + row) * Kdim + c4,
                                sx + (size_t)(t * 16 + row) * LDSS + c4);
            }
        }
        wait_async0();
    }
    __syncthreads();

    // ---- LayerNorm in place: 16 threads per row, shuffle-reduce width 16 ----
    {
        const int row = tid >> 4;
        const int t16 = tid & 15;
        for (int t = 0; t < NT; t++) {
            float* sr = sx + (size_t)(t * 16 + row) * LDSS;
            float s = 0.f;
            for (int col = t16; col < Kdim; col += 16) s += sr[col];
            for (int m = 8; m; m >>= 1) s += __shfl_xor(s, m, 16);
            const float mean = s / (float)Kdim;

            float vs = 0.f;
            for (int col = t16; col < Kdim; col += 16) {
                float d = sr[col] - mean;
                vs += d * d;
            }
            for (int m = 8; m; m >>= 1) vs += __shfl_xor(vs, m, 16);
            const float rstd = rsqrtf(vs / (float)Kdim + 1e-5f);

            for (int col = t16; col < Kdim; col += 16)
                sr[col] = (sr[col] - mean) * rstd * g[col] + b[col];
        }
    }
    __syncthreads();

    // ---- WMMA: k outermost, B fragment shared across NT tile accumulators ----
    const int w    = tid >> 5;        // wave id -> N tile
    const int lane = tid & 31;
    const int half = lane >> 4;       // 0: lanes 0-15, 1: lanes 16-31
    const int ln   = lane & 15;
    const int nb   = w * 16;

    v8f cc[NT];
    #pragma unroll
    for (int t = 0; t < NT; t++) cc[t] = (v8f){};

    for (int k = 0; k < Kdim; k += 4) {
        // B 4x16 f32: VGPR i holds row K = k + i + 2*half, N striped over lanes
        const float* brow = Wt + (size_t)(k + 2 * half) * DOUT + nb + ln;
        v2f bf;
        bf[0] = brow[0];
        bf[1] = brow[DOUT];
        #pragma unroll
        for (int t = 0; t < NT; t++) {
            // A 16x4 f32: VGPR i holds K = k + i + 2*half; row m = ln
            const float* arow = sx + (size_t)(t * 16 + ln) * LDSS + k + 2 * half;
            v2f a;
            a[0] = arow[0];
            a[1] = arow[1];
            cc[t] = __builtin_amdgcn_wmma_f32_16x16x4_f32(
                        false, a, false, bf, (short)0, cc[t], false, false);
        }
    }

    // ---- Epilogue: bias + SiLU; C/D layout: VGPR i -> M = i + 8*half, N = ln ----
    const float bias = wb[nb + ln];
    #pragma unroll
    for (int t = 0; t < NT; t++) {
        float* yb = y + (size_t)(row0 + t * 16) * DOUT + nb + ln;
        #pragma unroll
        for (int i = 0; i < 8; i++) {
            const int m = i + 8 * half;
            float v = cc[t][i] + bias;
            yb[(size_t)m * DOUT] = v / (1.f + __expf(-v));
        }
    }
}

// ---------------------------------------------------------------------------
// Phoneme occurrence counting
// ---------------------------------------------------------------------------
__global__ void zero_counts_kernel(int* counts, int V) {
    if (threadIdx.x < V) counts[threadIdx.x] = 0;
}

__global__ void count_kernel(const int* __restrict__ idx, int* __restrict__ counts, int K) {
    int g = blockIdx.x * blockDim.x + threadIdx.x;
    if (g < K) atomicAdd(&counts[idx[g]], 1);
}

// ---------------------------------------------------------------------------
// Closed-form sequential EMA: one block per phoneme slot v.
// ---------------------------------------------------------------------------
__global__ __launch_bounds__(128)
void ema_kernel(const int* __restrict__ idx, const float* __restrict__ xp,
                const float* __restrict__ ma, const int* __restrict__ counts,
                float* __restrict__ out, int K) {
    const int v    = blockIdx.x;
    const int tid  = threadIdx.x;
    const int lane = tid & 31;
    const int wid  = tid >> 5;

    __shared__ int slist[128];
    __shared__ int swtot[4];

    const int c = counts[v];
    float acc = 0.f;
    int r = 0;

    for (int base = 0; base < K; base += 128) {
        const int pos  = base + tid;
        const bool flg = (idx[pos] == v);
        const unsigned long long m = __ballot(flg);
        const int pfx = __popcll(m & ((1ull << lane) - 1ull));
        if (lane == 0) swtot[wid] = __popcll(m);
        __syncthreads();
        int wbase = 0, total = 0;
        #pragma unroll
        for (int i = 0; i < 4; i++) {
            int t2 = swtot[i];
            if (i < wid) wbase += t2;
            total += t2;
        }
        if (flg) slist[wbase + pfx] = pos;
        __syncthreads();
        for (int j = 0; j < total; j++) {
            const int p = slist[j];
            const float wgt = THTA * exp2f(L2_02 * (float)(c - 1 - r));
            acc += wgt * xp[(size_t)p * DOUT + tid];
            r++;
        }
        __syncthreads();
    }
    out[(size_t)v * DOUT + tid] =
        exp2f(L2_02 * (float)c) * ma[(size_t)v * DOUT + tid] + acc;
}

// ---------------------------------------------------------------------------
// Host launcher
// ---------------------------------------------------------------------------
extern "C" void kernel_launch(void* const* d_in, const int* in_sizes, int n_in,
                              void* d_out, int out_size, void* d_ws, size_t ws_size,
                              hipStream_t stream) {
    const int B = 64, T = 1000, Din = 384, D = 128, K = 131072, V = 70;
    const int M = B * T;                    // 64000 speech rows
    const size_t lds_bytes = (size_t)NT * 16 * LDSS * sizeof(float);  // ~97 KB

    const float* anco   = (const float*)d_in[0];
    const float* coms   = (const float*)d_in[1];
    const float* phnfea = (const float*)d_in[2];
    const float* myMA   = (const float*)d_in[3];
    const float* an_g   = (const float*)d_in[4];
    const float* an_b   = (const float*)d_in[5];
    const float* an_W   = (const float*)d_in[6];
    const float* an_wb  = (const float*)d_in[7];
    const float* com_g  = (const float*)d_in[8];
    const float* com_b  = (const float*)d_in[9];
    const float* com_W  = (const float*)d_in[10];
    const float* com_wb = (const float*)d_in[11];
    const float* an2_g  = (const float*)d_in[12];
    const float* an2_b  = (const float*)d_in[13];
    const float* an2_W  = (const float*)d_in[14];
    const float* an2_wb = (const float*)d_in[15];
    const float* com2_g  = (const float*)d_in[16];
    const float* com2_b  = (const float*)d_in[17];
    const float* com2_W  = (const float*)d_in[18];
    const float* com2_wb = (const float*)d_in[19];
    const int*   phn     = (const int*)d_in[20];

    // workspace carve-out (floats)
    float* ws       = (float*)d_ws;
    float* Wt_an    = ws;                               // 384*128
    float* Wt_com   = Wt_an   + (size_t)Din * D;
    float* Wt2_an   = Wt_com  + (size_t)Din * D;        // 128*128
    float* Wt2_com  = Wt2_an  + (size_t)D * D;
    float* anco_p   = Wt2_com + (size_t)D * D;          // M*128
    float* com_p    = anco_p  + (size_t)M * D;
    float* com_p2   = com_p   + (size_t)M * D;          // K*128
    int*   counts   = (int*)(com_p2 + (size_t)K * D);

    float* out      = (float*)d_out;
    float* anco_out = out;
    float* com_out  = out + (size_t)M * D;
    float* ma_new   = out + 2 * (size_t)M * D;

    // weight transposes (tiny)
    transpose_kernel<<<(D * Din + 255) / 256, 256, 0, stream>>>(an_W,   Wt_an,   D, Din);
    transpose_kernel<<<(D * Din + 255) / 256, 256, 0, stream>>>(com_W,  Wt_com,  D, Din);
    transpose_kernel<<<(D * D   + 255) / 256, 256, 0, stream>>>(an2_W,  Wt2_an,  D, D);
    transpose_kernel<<<(D * D   + 255) / 256, 256, 0, stream>>>(com2_W, Wt2_com, D, D);

    // stage-1 adapters (384 -> 128)
    stage_kernel<<<M / (16 * NT), 256, lds_bytes, stream>>>(anco, an_g,  an_b,  Wt_an,  an_wb,  anco_p, Din);
    stage_kernel<<<M / (16 * NT), 256, lds_bytes, stream>>>(coms, com_g, com_b, Wt_com, com_wb, com_p,  Din);
    // pooled phoneme features through second com adapter (128 -> 128)
    stage_kernel<<<K / (16 * NT), 256, lds_bytes, stream>>>(phnfea, com2_g, com2_b, Wt2_com, com2_wb, com_p2, D);
    // stage-2 adapters (128 -> 128) straight into d_out
    stage_kernel<<<M / (16 * NT), 256, lds_bytes, stream>>>(anco_p, an2_g,  an2_b,  Wt2_an,  an2_wb,  anco_out, D);
    stage_kernel<<<M / (16 * NT), 256, lds_bytes, stream>>>(com_p,  com2_g, com2_b, Wt2_com, com2_wb, com_out,  D);

    // EMA update
    zero_counts_kernel<<<1, 128, 0, stream>>>(counts, V);
    count_kernel<<<(K + 255) / 256, 256, 0, stream>>>(phn, counts, K);
    ema_kernel<<<V, 128, 0, stream>>>(phn, com_p2, myMA, counts, ma_new, K);
}